// AttentionCNN_38122129719492
// MI455X (gfx1250) — compile-verified
//
#include <hip/hip_runtime.h>
#include <hip/hip_bf16.h>
#include <math.h>

// ---------------------------------------------------------------------------
// AttentionCNN on MI455X (gfx1250, wave32).
// GEMM work -> V_WMMA_F32_16X16X4_F32.
// fc1 (1 GiB fp32 weight stream, HBM-bound, ~46us floor @ 23.3 TB/s) streams
// weight tiles with the Tensor Data Mover (TENSOR_LOAD_TO_LDS) double-buffered
// against WMMA compute; D# pad fields give a conflict-free LDS stride of 68.
// ---------------------------------------------------------------------------

#define Bsz   16
#define Cch   64
#define CCch  16
#define NPIX  1024            // H*W
#define FCIN  65536           // Cch*NPIX
#define FC1N  4096

typedef float v2f __attribute__((ext_vector_type(2)));
typedef float v8f __attribute__((ext_vector_type(8)));
typedef float f32x4 __attribute__((ext_vector_type(4)));
typedef unsigned int u32x4 __attribute__((ext_vector_type(4)));
typedef int i32x4 __attribute__((ext_vector_type(4)));
typedef int i32x8 __attribute__((ext_vector_type(8)));

__device__ __forceinline__ v8f wmma_f32(v2f a, v2f b, v8f c) {
  return __builtin_amdgcn_wmma_f32_16x16x4_f32(false, a, false, b, (short)0, c,
                                               false, false);
}

// ---------------------------------------------------------------------------
// 3x3 conv, 1 input channel -> 64 output channels (zero pad).
// ---------------------------------------------------------------------------
__global__ void conv3x3_c1(const float* __restrict__ x, int slice,
                           const float* __restrict__ w,
                           const float* __restrict__ bias,
                           float* __restrict__ out) {
  int idx = blockIdx.x * blockDim.x + threadIdx.x;   // B*C*N threads
  int p  = idx & (NPIX - 1);
  int co = (idx >> 10) & (Cch - 1);
  int b  = idx >> 16;
  int py = p >> 5, px = p & 31;
  const float* xin = x + ((size_t)(b * 4 + slice)) * NPIX;
  const float* wk  = w + co * 9;
  float acc = bias[co];
#pragma unroll
  for (int ky = 0; ky < 3; ++ky) {
    int y = py + ky - 1;
    if ((unsigned)y >= 32u) continue;
#pragma unroll
    for (int kx = 0; kx < 3; ++kx) {
      int xx = px + kx - 1;
      if ((unsigned)xx >= 32u) continue;
      acc += wk[ky * 3 + kx] * xin[y * 32 + xx];
    }
  }
  out[idx] = acc;
}

// ---------------------------------------------------------------------------
// 3x3 conv, 64 -> 64 channels (convf).
// ---------------------------------------------------------------------------
__global__ void conv3x3_cc(const float* __restrict__ x,
                           const float* __restrict__ w,
                           const float* __restrict__ bias,
                           float* __restrict__ out) {
  int idx = blockIdx.x * blockDim.x + threadIdx.x;
  int p  = idx & (NPIX - 1);
  int co = (idx >> 10) & (Cch - 1);
  int b  = idx >> 16;
  int py = p >> 5, px = p & 31;
  float acc = bias[co];
  const float* xb = x + (size_t)b * Cch * NPIX;
  const float* wc = w + (size_t)co * Cch * 9;
  for (int ci = 0; ci < Cch; ++ci) {
    const float* xc = xb + ci * NPIX;
    const float* wk = wc + ci * 9;
#pragma unroll
    for (int ky = 0; ky < 3; ++ky) {
      int y = py + ky - 1;
      if ((unsigned)y >= 32u) continue;
#pragma unroll
      for (int kx = 0; kx < 3; ++kx) {
        int xx = px + kx - 1;
        if ((unsigned)xx >= 32u) continue;
        acc += wk[ky * 3 + kx] * xc[y * 32 + xx];
      }
    }
  }
  out[idx] = acc;
}

// ---------------------------------------------------------------------------
// Pointwise conv (1x1) as WMMA GEMM: Y[b] = W[Co x Ci] * X[b][Ci x N] + bias.
// One wave per 16(co) x 16(n) output tile; grid sized to exact full waves.
// ---------------------------------------------------------------------------
__global__ void pw_wmma(const float* __restrict__ W,
                        const float* __restrict__ bias,
                        const float* __restrict__ X, float* __restrict__ Y,
                        int Ci, int Co) {
  int wid  = (blockIdx.x * blockDim.x + threadIdx.x) >> 5;
  int lane = threadIdx.x & 31;
  int half = lane >> 4, l16 = lane & 15;
  const int ntn = NPIX / 16;                 // 64 n-tiles
  int cot = wid / ntn, nt = wid % ntn;
  int b   = blockIdx.y;
  const float* Xb = X + (size_t)b * Ci * NPIX;
  float*       Yb = Y + (size_t)b * Co * NPIX;
  int co0 = cot * 16, n0 = nt * 16;
  v8f acc = {};
  for (int k = 0; k < Ci; k += 4) {
    int c0 = k + 2 * half;
    v2f a, bm;
    a.x  = W[(co0 + l16) * Ci + c0];
    a.y  = W[(co0 + l16) * Ci + c0 + 1];
    bm.x = Xb[(size_t)c0 * NPIX + n0 + l16];
    bm.y = Xb[(size_t)(c0 + 1) * NPIX + n0 + l16];
    acc = wmma_f32(a, bm, acc);
  }
#pragma unroll
  for (int v = 0; v < 8; ++v) {
    int row = co0 + v + 8 * half;
    Yb[(size_t)row * NPIX + n0 + l16] = acc[v] + bias[row];
  }
}

// ---------------------------------------------------------------------------
// Attention logits: attn[b][n][m] = scale * sum_c Q[b][c][n] * K[b][c][m].
// ---------------------------------------------------------------------------
__global__ void attn_logits_wmma(const float* __restrict__ Q,
                                 const float* __restrict__ Kt,
                                 float* __restrict__ attn, int Cdim,
                                 float scale) {
  int wid  = (blockIdx.x * blockDim.x + threadIdx.x) >> 5;
  int lane = threadIdx.x & 31;
  int half = lane >> 4, l16 = lane & 15;
  int nt = wid >> 6, mt = wid & 63;           // 64 x 64 tiles
  int b  = blockIdx.y;
  const float* Qb = Q + (size_t)b * Cdim * NPIX;
  const float* Kb = Kt + (size_t)b * Cdim * NPIX;
  int n0 = nt * 16, m0 = mt * 16;
  v8f acc = {};
  for (int k = 0; k < Cdim; k += 4) {
    int c0 = k + 2 * half;
    v2f a, bm;
    a.x  = Qb[(size_t)c0 * NPIX + n0 + l16];
    a.y  = Qb[(size_t)(c0 + 1) * NPIX + n0 + l16];
    bm.x = Kb[(size_t)c0 * NPIX + m0 + l16];
    bm.y = Kb[(size_t)(c0 + 1) * NPIX + m0 + l16];
    acc = wmma_f32(a, bm, acc);
  }
  float* ab = attn + (size_t)b * NPIX * NPIX;
#pragma unroll
  for (int v = 0; v < 8; ++v) {
    int n = n0 + v + 8 * half;
    ab[(size_t)n * NPIX + m0 + l16] = acc[v] * scale;
  }
}

// ---------------------------------------------------------------------------
// Row softmax over 1024 elements; one block (256 threads) per row.
// ---------------------------------------------------------------------------
__global__ void softmax_rows(float* __restrict__ attn) {
  __shared__ float red[256];
  float* p = attn + (size_t)blockIdx.x * NPIX;
  int t = threadIdx.x;
  float m = -INFINITY;
  for (int i = t; i < NPIX; i += 256) m = fmaxf(m, p[i]);
  red[t] = m;
  __syncthreads();
  for (int s = 128; s > 0; s >>= 1) {
    if (t < s) red[t] = fmaxf(red[t], red[t + s]);
    __syncthreads();
  }
  m = red[0];
  __syncthreads();
  float sum = 0.0f;
  for (int i = t; i < NPIX; i += 256) {
    float e = __expf(p[i] - m);
    p[i] = e;
    sum += e;
  }
  red[t] = sum;
  __syncthreads();
  for (int s = 128; s > 0; s >>= 1) {
    if (t < s) red[t] += red[t + s];
    __syncthreads();
  }
  float inv = 1.0f / red[0];
  for (int i = t; i < NPIX; i += 256) p[i] *= inv;
}

// ---------------------------------------------------------------------------
// Apply attention: out[b][c][n] = sum_m attn[b][n][m]*V[b][c][m] + res[b][c][n]
// ---------------------------------------------------------------------------
__global__ void attn_apply_wmma(const float* __restrict__ attn,
                                const float* __restrict__ V,
                                const float* __restrict__ res,
                                float* __restrict__ out, int Cdim) {
  int wid  = (blockIdx.x * blockDim.x + threadIdx.x) >> 5;
  int lane = threadIdx.x & 31;
  int half = lane >> 4, l16 = lane & 15;
  int ct = wid / 64, nt = wid % 64;
  int b  = blockIdx.y;
  const float* Vb = V + (size_t)b * Cdim * NPIX;
  const float* ab = attn + (size_t)b * NPIX * NPIX;
  int c0 = ct * 16, n0 = nt * 16;
  v8f acc = {};
  for (int k = 0; k < NPIX; k += 4) {
    int m0 = k + 2 * half;
    v2f a, bm;
    a.x  = Vb[(size_t)(c0 + l16) * NPIX + m0];
    a.y  = Vb[(size_t)(c0 + l16) * NPIX + m0 + 1];
    bm.x = ab[(size_t)(n0 + l16) * NPIX + m0];
    bm.y = ab[(size_t)(n0 + l16) * NPIX + m0 + 1];
    acc = wmma_f32(a, bm, acc);
  }
  const float* rb = res + (size_t)b * Cdim * NPIX;
  float*       ob = out + (size_t)b * Cdim * NPIX;
#pragma unroll
  for (int v = 0; v < 8; ++v) {
    int c = c0 + v + 8 * half;
    ob[(size_t)c * NPIX + n0 + l16] =
        acc[v] + rb[(size_t)c * NPIX + n0 + l16];
  }
}

// ---------------------------------------------------------------------------
// fc1: h1acc[b][j] += sum_k W[j][k] * X[b][k]   (K-split, f32 atomics)
// Tile: 128 neurons x 16 batch; KB=64 K-chunk. Weight tiles stream into LDS
// via the Tensor Data Mover, double-buffered against WMMA; D# pad fields
// (4 DWORDs every 64 DWORDs) realize the bank-conflict-free stride of 68.
// ---------------------------------------------------------------------------
#define FC1_KSPLIT   16
#define FC1_KB       64
#define FC1_ROWS     128
#define FC1_LDSTRIDE 68
#define FC1_ITERS    ((FCIN / FC1_KSPLIT) / FC1_KB)   // 64

#if __has_builtin(__builtin_amdgcn_tensor_load_to_lds) && \
    __has_builtin(__builtin_amdgcn_s_wait_tensorcnt)
#define FC1_USE_TDM 1
#else
#define FC1_USE_TDM 0
#endif

__global__ void fc1_wmma(const float* __restrict__ W,
                         const float* __restrict__ X,
                         float* __restrict__ acc_out) {
  __shared__ float Ws[2][FC1_ROWS][FC1_LDSTRIDE];
  __shared__ float Xs[16][FC1_LDSTRIDE];
  int nblk = blockIdx.x;                     // 32 blocks of 128 neurons
  int kblk = blockIdx.y;                     // K split
  int t = threadIdx.x;
  int wave = t >> 5, lane = t & 31, half = lane >> 4, l16 = lane & 15;
  int n0 = nblk * FC1_ROWS;
  int kbase = kblk * (FCIN / FC1_KSPLIT);    // 4096 per split
  v8f acc = {};

#if FC1_USE_TDM
  // Issue a TDM load of tile W[n0 .. n0+127][k0 .. k0+63] into Ws[buf].
  // D# group0: count=1 | lds_addr | global_addr | type=2 ("image").
  // D# group1: data_size=4B, pad_enable, pad_interval=5 (64 DW),
  //            pad_amount=3 (4 DW) -> LDS row stride 68 floats;
  //            tensor 65536 x 4096, tile 64 x 128, dim0_stride 65536.
  auto tdm_issue = [&](int buf, int k0) {
    unsigned long long ga =
        (unsigned long long)(const void*)(W + (size_t)n0 * FCIN + k0);
    u32x4 g0;
    g0.x = 1u;                                                  // count=1
    g0.y = (unsigned)(buf * (FC1_ROWS * FC1_LDSTRIDE * 4));     // lds_addr
    g0.z = (unsigned)(ga & 0xffffffffu);
    g0.w = (unsigned)((ga >> 32) & 0x1ffffffu) | (2u << 30);    // type=2
    i32x8 g1;
    g1[0] = (int)((2u << 16) | (1u << 20) | (5u << 22) | (3u << 25));
    g1[1] = 0;                               // barrier addr 0, dim0 lo16 = 0
    g1[2] = (int)((0x1000u << 16) | 1u);     // dim0 hi16 = 1, dim1 = 4096
    g1[3] = (int)(64u << 16);                // tile_dim0 = 64
    g1[4] = 128;                             // tile_dim1 = 128
    g1[5] = 65536;                           // dim0_stride lo32
    g1[6] = 0;
    g1[7] = 0;
    i32x4 z4 = {0, 0, 0, 0};                 // groups 2/3 unused (2-D)
    i32x8 z8 = {0, 0, 0, 0, 0, 0, 0, 0};     // trailing group (clang-23 form)
    __builtin_amdgcn_tensor_load_to_lds(g0, g1, z4, z4, z8, 0);
  };

  if (t < 32) tdm_issue(0, kbase);
  for (int it = 0; it < FC1_ITERS; ++it) {
    int k0 = kbase + it * FC1_KB;
    if (t < 32) {
      if (it + 1 < FC1_ITERS) {
        tdm_issue((it + 1) & 1, kbase + (it + 1) * FC1_KB);
        __builtin_amdgcn_s_wait_tensorcnt(1);   // current buffer landed
      } else {
        __builtin_amdgcn_s_wait_tensorcnt(0);
      }
    }
    // Stage X tile 16x64: one float4 per thread (L2-resident activations).
    {
      int row  = t >> 4;
      int col4 = (t & 15) * 4;
      f32x4 v = *(const f32x4*)(X + (size_t)row * FCIN + k0 + col4);
      Xs[row][col4 + 0] = v.x;
      Xs[row][col4 + 1] = v.y;
      Xs[row][col4 + 2] = v.z;
      Xs[row][col4 + 3] = v.w;
    }
    __syncthreads();
    int wr = wave * 16;
    int buf = it & 1;
#pragma unroll
    for (int kk = 0; kk < FC1_KB; kk += 4) {
      int c0 = kk + 2 * half;
      v2f a, bm;
      a.x  = Ws[buf][wr + l16][c0];
      a.y  = Ws[buf][wr + l16][c0 + 1];
      bm.x = Xs[l16][c0];
      bm.y = Xs[l16][c0 + 1];
      acc = wmma_f32(a, bm, acc);
    }
    __syncthreads();
  }
#else
  // Fallback: synchronous staging (nontemporal: weights are stream-once).
  for (int it = 0; it < FC1_ITERS; ++it) {
    int k0 = kbase + it * FC1_KB;
    {
      int r = t >> 1;
      int cpart = (t & 1) * 32;
      const float* src = W + (size_t)(n0 + r) * FCIN + k0 + cpart;
#pragma unroll
      for (int i = 0; i < 8; ++i) {
        f32x4 v = __builtin_nontemporal_load((const f32x4*)src + i);
        Ws[0][r][cpart + 4 * i + 0] = v.x;
        Ws[0][r][cpart + 4 * i + 1] = v.y;
        Ws[0][r][cpart + 4 * i + 2] = v.z;
        Ws[0][r][cpart + 4 * i + 3] = v.w;
      }
    }
    {
      int row  = t >> 4;
      int col4 = (t & 15) * 4;
      f32x4 v = *(const f32x4*)(X + (size_t)row * FCIN + k0 + col4);
      Xs[row][col4 + 0] = v.x;
      Xs[row][col4 + 1] = v.y;
      Xs[row][col4 + 2] = v.z;
      Xs[row][col4 + 3] = v.w;
    }
    __syncthreads();
    int wr = wave * 16;
#pragma unroll
    for (int kk = 0; kk < FC1_KB; kk += 4) {
      int c0 = kk + 2 * half;
      v2f a, bm;
      a.x  = Ws[0][wr + l16][c0];
      a.y  = Ws[0][wr + l16][c0 + 1];
      bm.x = Xs[l16][c0];
      bm.y = Xs[l16][c0 + 1];
      acc = wmma_f32(a, bm, acc);
    }
    __syncthreads();
  }
#endif

  // D[m=neuron][n=batch] -> atomic partial reduction across K splits.
#pragma unroll
  for (int v = 0; v < 8; ++v) {
    int neuron = n0 + wave * 16 + v + 8 * half;
    atomicAdd(&acc_out[(size_t)l16 * FC1N + neuron], acc[v]);
  }
}

__global__ void zero_kernel(float* __restrict__ p, int n) {
  int i = blockIdx.x * blockDim.x + threadIdx.x;
  if (i < n) p[i] = 0.0f;
}

__global__ void bias_relu(float* __restrict__ h, const float* __restrict__ bias,
                          int n) {
  int idx = blockIdx.x * blockDim.x + threadIdx.x;
  int j = idx & (n - 1);
  h[idx] = fmaxf(h[idx] + bias[j], 0.0f);
}

// fc2: one wave per output (b,j), wave-reduced dot of 4096.
__global__ void fc2_kernel(const float* __restrict__ h1,
                           const float* __restrict__ w,
                           const float* __restrict__ bias,
                           float* __restrict__ h2) {
  int wid  = (blockIdx.x * blockDim.x + threadIdx.x) >> 5;
  int lane = threadIdx.x & 31;
  if (wid >= Bsz * 128) return;
  int b = wid >> 7, j = wid & 127;
  const float* hr = h1 + (size_t)b * FC1N;
  const float* wr = w + (size_t)j * FC1N;
  float s = 0.0f;
  for (int k = lane; k < FC1N; k += 32) s += hr[k] * wr[k];
  for (int off = 16; off > 0; off >>= 1) s += __shfl_down(s, off, 32);
  if (lane == 0) h2[b * 128 + j] = fmaxf(s + bias[j], 0.0f);
}

// fc3: one wave per output (b,j), j in {0,1}.
__global__ void fc3_kernel(const float* __restrict__ h2,
                           const float* __restrict__ w,
                           const float* __restrict__ bias,
                           float* __restrict__ out) {
  int wid  = (blockIdx.x * blockDim.x + threadIdx.x) >> 5;
  int lane = threadIdx.x & 31;
  if (wid >= Bsz * 2) return;
  int b = wid >> 1, j = wid & 1;
  const float* hr = h2 + b * 128;
  const float* wr = w + j * 128;
  float s = 0.0f;
  for (int k = lane; k < 128; k += 32) s += hr[k] * wr[k];
  for (int off = 16; off > 0; off >>= 1) s += __shfl_down(s, off, 32);
  if (lane == 0) out[b * 2 + j] = s + bias[j];
}

// ---------------------------------------------------------------------------
// Host-side orchestration.
// ---------------------------------------------------------------------------
extern "C" void kernel_launch(void* const* d_in, const int* in_sizes, int n_in,
                              void* d_out, int out_size, void* d_ws,
                              size_t ws_size, hipStream_t stream) {
  const float* x    = (const float*)d_in[0];
  const float* c1w  = (const float*)d_in[1];
  const float* c1b  = (const float*)d_in[2];
  const float* c2w  = (const float*)d_in[3];
  const float* c2b  = (const float*)d_in[4];
  const float* c3w  = (const float*)d_in[5];
  const float* c3b  = (const float*)d_in[6];
  const float* cfw  = (const float*)d_in[7];
  const float* cfb  = (const float*)d_in[8];
  const float* qw   = (const float*)d_in[9];
  const float* qb   = (const float*)d_in[10];
  const float* kw   = (const float*)d_in[11];
  const float* kb   = (const float*)d_in[12];
  const float* vw   = (const float*)d_in[13];
  const float* vb   = (const float*)d_in[14];
  const float* ow   = (const float*)d_in[15];
  const float* ob   = (const float*)d_in[16];
  const float* fc1w = (const float*)d_in[35];
  const float* fc1b = (const float*)d_in[36];
  const float* fc2w = (const float*)d_in[37];
  const float* fc2b = (const float*)d_in[38];
  const float* fc3w = (const float*)d_in[39];
  const float* fc3b = (const float*)d_in[40];

  float* base = (float*)d_ws;
  size_t off = 0;
  auto take = [&](size_t n) { float* p = base + off; off += n; return p; };

  const size_t BCN  = (size_t)Bsz * Cch * NPIX;    // 1,048,576
  const size_t BCCN = (size_t)Bsz * CCch * NPIX;   //   262,144

  float* F1   = take(BCN);
  float* F2   = take(BCN);
  float* F3   = take(BCN);
  float* F4   = take(BCN);
  float* Qb_  = take(BCN);
  float* Kb_  = take(BCN);
  float* Vb_  = take(BCN);
  float* Tb   = take(BCN);   // self-attn output / pw input
  float* Ab0  = take(BCN);   // A1, A3
  float* Ab1  = take(BCN);   // A2, A4
  float* Fc   = take(BCCN);
  float* Ac   = take(BCCN);
  float* AttS = take(BCCN);
  float* feat = take(BCN);   // flattened view matches [B][65536]
  float* h1   = take((size_t)Bsz * FC1N);
  float* h2   = take((size_t)Bsz * 128);
  float* attn = take((size_t)Bsz * NPIX * NPIX);   // 64 MB
  (void)ws_size; (void)in_sizes; (void)n_in; (void)out_size;

  const int CONV_BLOCKS = (int)(BCN / 256);        // 4096

  // --- F1..F4 (F1 and F2 both use conv1 weights, per reference) ---
  conv3x3_c1<<<CONV_BLOCKS, 256, 0, stream>>>(x, 0, c1w, c1b, F1);
  conv3x3_c1<<<CONV_BLOCKS, 256, 0, stream>>>(x, 1, c1w, c1b, F2);
  conv3x3_c1<<<CONV_BLOCKS, 256, 0, stream>>>(x, 2, c2w, c2b, F3);
  conv3x3_c1<<<CONV_BLOCKS, 256, 0, stream>>>(x, 3, c3w, c3b, F4);

  // --- self attention on F1 -> A1 (Ab0) ---
  pw_wmma<<<dim3(32, Bsz), 256, 0, stream>>>(qw, qb, F1, Qb_, Cch, Cch);
  pw_wmma<<<dim3(32, Bsz), 256, 0, stream>>>(kw, kb, F1, Kb_, Cch, Cch);
  pw_wmma<<<dim3(32, Bsz), 256, 0, stream>>>(vw, vb, F1, Vb_, Cch, Cch);
  attn_logits_wmma<<<dim3(512, Bsz), 256, 0, stream>>>(Qb_, Kb_, attn, Cch,
                                                       0.125f);  // 1/sqrt(64)
  softmax_rows<<<Bsz * NPIX, 256, 0, stream>>>(attn);
  attn_apply_wmma<<<dim3(32, Bsz), 256, 0, stream>>>(attn, Vb_, Vb_, Tb, Cch);
  pw_wmma<<<dim3(32, Bsz), 256, 0, stream>>>(ow, ob, Tb, Ab0, Cch, Cch);

  // --- multi-scale attention stages ---
  auto msa = [&](const float* Aprev, const float* Fn, int base_idx,
                 float* Aout) {
    const float* fnw = (const float*)d_in[base_idx + 0];
    const float* fnb = (const float*)d_in[base_idx + 1];
    const float* anw = (const float*)d_in[base_idx + 2];
    const float* anb = (const float*)d_in[base_idx + 3];
    const float* ow_ = (const float*)d_in[base_idx + 4];
    const float* ob_ = (const float*)d_in[base_idx + 5];
    pw_wmma<<<dim3(8, Bsz), 256, 0, stream>>>(fnw, fnb, Fn, Fc, Cch, CCch);
    pw_wmma<<<dim3(8, Bsz), 256, 0, stream>>>(anw, anb, Aprev, Ac, Cch, CCch);
    attn_logits_wmma<<<dim3(512, Bsz), 256, 0, stream>>>(Fc, Ac, attn, CCch,
                                                         1.0f);
    softmax_rows<<<Bsz * NPIX, 256, 0, stream>>>(attn);
    attn_apply_wmma<<<dim3(8, Bsz), 256, 0, stream>>>(attn, Fc, Fc, AttS,
                                                      CCch);
    pw_wmma<<<dim3(32, Bsz), 256, 0, stream>>>(ow_, ob_, AttS, Aout, CCch,
                                               Cch);
  };
  msa(Ab0, F2, 17, Ab1);   // A2
  msa(Ab1, F3, 23, Ab0);   // A3
  msa(Ab0, F4, 29, Ab1);   // A4

  // --- convf -> feat (== flattened [B,65536]) ---
  conv3x3_cc<<<CONV_BLOCKS, 256, 0, stream>>>(Ab1, cfw, cfb, feat);

  // --- fc1 (HBM-bound 1 GiB weight stream: TDM + WMMA + K-split atomics) ---
  zero_kernel<<<(Bsz * FC1N) / 256, 256, 0, stream>>>(h1, Bsz * FC1N);
  fc1_wmma<<<dim3(FC1N / 128, FC1_KSPLIT), 256, 0, stream>>>(fc1w, feat, h1);
  bias_relu<<<(Bsz * FC1N) / 256, 256, 0, stream>>>(h1, fc1b, FC1N);

  // --- fc2 / fc3 ---
  fc2_kernel<<<(Bsz * 128 * 32) / 256, 256, 0, stream>>>(h1, fc2w, fc2b, h2);
  fc3_kernel<<<4, 256, 0, stream>>>(h2, fc3w, fc3b, (float*)d_out);
}